// Criterion_82738249990265
// MI455X (gfx1250) — compile-verified
//
#include <hip/hip_runtime.h>
#include <math.h>

// ---------------------------------------------------------------------------
// Shapes (fixed by the reference)
// ---------------------------------------------------------------------------
#define BB 8            // batch (videos == sentences)
#define CC 128          // channels
#define DD 64
#define TT 64
#define H1 66           // 64 + 2*FIRST_PAD - 3 + 1   (conv0 output side)
#define P1 (H1 * H1)    // 4356 positions after conv0
#define P0 (DD * TT)    // 4096 positions after conv1
#define KDIM (CC * 9)   // 1152 = GEMM K dimension
#define LW 20
#define MARGIN 0.1f
#define NEG_NUM 3

typedef float v2f __attribute__((ext_vector_type(2)));
typedef float v8f __attribute__((ext_vector_type(8)));

// ---------------------------------------------------------------------------
// Weight transpose: W[m][ci*9+tap] -> Wt[m][tap*128+ci]  (tap-major K order,
// so the GEMM inner loop advances ci with pure pointer increments)
// ---------------------------------------------------------------------------
__global__ void transpose_w_kernel(const float* __restrict__ W,
                                   float* __restrict__ Wt)
{
    int idx = blockIdx.x * 256 + threadIdx.x;      // over CC*KDIM
    if (idx < CC * KDIM) {
        int m  = idx / KDIM;
        int k  = idx - m * KDIM;
        int ci = k / 9;
        int tap = k - ci * 9;
        Wt[m * KDIM + tap * CC + ci] = W[idx];
    }
}

// ---------------------------------------------------------------------------
// prep: weight0 [66x66], weight1 [64x64], pooled->g0/g1 [8x128]  (one block)
// ---------------------------------------------------------------------------
__global__ void prep_kernel(const float* __restrict__ v_mask,
                            const float* __restrict__ words,
                            const float* __restrict__ w_masks,
                            const float* __restrict__ Wp0, const float* __restrict__ bp0,
                            const float* __restrict__ Wp1, const float* __restrict__ bp1,
                            float* __restrict__ weight0, float* __restrict__ weight1,
                            float* __restrict__ g0, float* __restrict__ g1)
{
    __shared__ float w0s[P1];
    __shared__ float pooled[BB * CC];
    const int tid = threadIdx.x;   // 256 threads

    for (int idx = tid; idx < P1; idx += 256) {
        int oy = idx / H1, ox = idx % H1;
        float s = 0.f;
        for (int ky = 0; ky < 3; ++ky)
            for (int kx = 0; kx < 3; ++kx) {
                int iy = oy + ky - 2, ix = ox + kx - 2;
                if (iy >= 0 && iy < DD && ix >= 0 && ix < TT)
                    s += v_mask[iy * TT + ix];
            }
        float w = (s > 0.f) ? 1.f / fmaxf(s, 1.f) : 0.f;
        w0s[idx] = w;
        weight0[idx] = w;
    }
    for (int idx = tid; idx < BB * CC; idx += 256) {
        int i = idx / CC, ch = idx % CC;
        float num = 0.f, den = 0.f;
        for (int l = 0; l < LW; ++l) {
            float wm = w_masks[i * LW + l];
            num += words[(i * LW + l) * CC + ch] * wm;
            den += wm;
        }
        pooled[idx] = num / fmaxf(den, 1.f);
    }
    __syncthreads();
    for (int idx = tid; idx < P0; idx += 256) {
        int oy = idx / TT, ox = idx % TT;
        float s = 0.f;
        for (int ky = 0; ky < 3; ++ky)
            for (int kx = 0; kx < 3; ++kx)
                s += (w0s[(oy + ky) * H1 + (ox + kx)] > 0.f) ? 1.f : 0.f;
        weight1[idx] = (s > 0.f) ? 1.f / fmaxf(s, 1.f) : 0.f;
    }
    for (int idx = tid; idx < BB * CC; idx += 256) {
        int i = idx / CC, j = idx % CC;
        float a0 = bp0[j], a1 = bp1[j];
        for (int k = 0; k < CC; ++k) {
            float pk = pooled[i * CC + k];
            a0 += pk * Wp0[k * CC + j];
            a1 += pk * Wp1[k * CC + j];
        }
        g0[idx] = tanhf(a0);
        g1[idx] = tanhf(a1);
    }
}

// ---------------------------------------------------------------------------
// conv0: implicit-GEMM with V_WMMA_F32_16X16X4_F32, tap-major K order.
// Branchless border handling: clamped address + cndmask select.
// out[v][cout][p] = conv2d(v_map, Wc0, pad=2) + bc0   (66x66 output)
// ---------------------------------------------------------------------------
__global__ void conv0_kernel(const float* __restrict__ v_map, // [BB][CC][64*64]
                             const float* __restrict__ Wt0,   // [CC][9*CC] tap-major
                             const float* __restrict__ bc0,   // [CC]
                             float* __restrict__ out)         // [BB][CC][P1]
{
    const int lane = threadIdx.x & 31;
    const int wav  = threadIdx.x >> 5;
    const int nPT  = (P1 + 15) / 16;     // 273
    const int pt   = blockIdx.x % nPT;
    const int v    = blockIdx.x / nPT;

    const int n  = lane & 15;
    const int p  = pt * 16 + n;
    const bool pOK = (p < P1);
    const int pc = pOK ? p : 0;
    const int oy = pc / H1;
    const int ox = pc - oy * H1;
    const int khalf = (lane >> 4) * 2;   // lanes 0-15: K0/K1, lanes 16-31: K2/K3
    const int m  = wav * 16 + n;         // cout row of A

    const float* Arow = Wt0 + (size_t)m * KDIM;
    const float* vsrc = v_map + (size_t)v * CC * P0;

    v8f acc = {};
#pragma unroll
    for (int tap = 0; tap < 9; ++tap) {
        const int ky = tap / 3, kx = tap % 3;        // constants after unroll
        const int iy = oy + ky - 2;
        const int ix = ox + kx - 2;
        const bool ok = pOK && (iy >= 0) && (iy < DD) && (ix >= 0) && (ix < TT);
        const int iyc = min(max(iy, 0), DD - 1);
        const int ixc = min(max(ix, 0), TT - 1);
        const float* bbase = vsrc + ((size_t)(khalf) * P0) + iyc * TT + ixc;
        const float* abase = Arow + tap * CC + khalf;
#pragma unroll 8
        for (int c4 = 0; c4 < CC; c4 += 4) {
            v2f a, b;
            a.x = abase[c4 + 0];
            a.y = abase[c4 + 1];
            float v0 = bbase[(size_t)c4 * P0];
            float v1 = bbase[(size_t)c4 * P0 + P0];
            b.x = ok ? v0 : 0.f;
            b.y = ok ? v1 : 0.f;
            acc = __builtin_amdgcn_wmma_f32_16x16x4_f32(false, a, false, b,
                                                        (short)0, acc, false, false);
        }
    }
    if (pOK) {
        const int rowAdd = 8 * (lane >> 4);
#pragma unroll
        for (int vv = 0; vv < 8; ++vv) {
            int cout = wav * 16 + vv + rowAdd;
            out[((size_t)v * CC + cout) * P1 + p] = acc[vv] + bc0[cout];
        }
    }
}

// ---------------------------------------------------------------------------
// conv1 + gating + 1x1 scoring conv + sigmoid + mask, fused.
// Per (sentence i, video v, 16-pos tile): the gated im2col slice is shared by
// all 8 waves, so it is staged per-tap in LDS (128 ci x 16 px = 8 KB) with
// weight0/g0/relu applied at stage time. All 16 positions lie in one 64-wide
// row (16 | 64), so addresses are shift/mask only.
// ---------------------------------------------------------------------------
__global__ void conv1_score_kernel(const float* __restrict__ conv0,   // [BB][CC][P1]
                                   const float* __restrict__ Wt1,     // [CC][9*CC]
                                   const float* __restrict__ bc1,     // [CC]
                                   const float* __restrict__ weight0, // [P1]
                                   const float* __restrict__ weight1, // [P0]
                                   const float* __restrict__ g0,      // [BB][CC]
                                   const float* __restrict__ g1,      // [BB][CC]
                                   const float* __restrict__ W2d,     // [CC]
                                   const float* __restrict__ b2d,     // [1]
                                   const float* __restrict__ v_mask,  // [P0]
                                   const int*   __restrict__ valid_num,
                                   float* __restrict__ s)             // [BB][BB][P0]
{
    __shared__ float xs[CC * 16];        // 8 KB per-tap gated B slice
    __shared__ float red[256];

    const int lane = threadIdx.x & 31;
    const int wav  = threadIdx.x >> 5;
    const int bid  = blockIdx.x;
    const int pt   = bid & 255;          // 256 pos tiles
    const int v    = (bid >> 8) & 7;     // video
    const int i    = bid >> 11;          // sentence

    const int oy  = pt >> 2;             // tile row (0..63)
    const int ox0 = (pt & 3) << 4;       // tile col start
    const int n   = lane & 15;
    const int p   = (pt << 4) + n;       // 0..4095
    const int khalf = (lane >> 4) * 2;
    const int m   = wav * 16 + n;

    const float* Arow = Wt1 + (size_t)m * KDIM;
    const float* src  = conv0 + (size_t)v * CC * P1;

    // staging role: thread -> (ci, pixel half)
    const int   tci   = threadIdx.x & 127;
    const int   thalf = threadIdx.x >> 7;          // pixels [8h, 8h+8)
    const float gci   = g0[i * CC + tci];
    const float* srcc = src + (size_t)tci * P1;

    v8f acc = {};
#pragma unroll
    for (int tap = 0; tap < 9; ++tap) {
        const int ky = tap / 3, kx = tap % 3;      // constants after unroll
        // ---- cooperative stage: xs[ci][px] = relu(conv0 * weight0 * g0) ----
        {
            const int row66 = (oy + ky) * H1 + ox0 + kx + thalf * 8;
            const float* gsrc = srcc + row66;
            const float* w0p  = weight0 + row66;
            float* dst = xs + tci * 16 + thalf * 8;
#pragma unroll
            for (int j = 0; j < 8; ++j)
                dst[j] = fmaxf(gsrc[j] * w0p[j] * gci, 0.f);
        }
        __syncthreads();
        // ---- 32 WMMA steps over this tap's 128 ci ----
        const float* abase = Arow + tap * CC + khalf;
        const float* bbase = xs + khalf * 16 + n;
#pragma unroll 8
        for (int c4 = 0; c4 < CC; c4 += 4) {
            v2f a, b;
            a.x = abase[c4 + 0];
            a.y = abase[c4 + 1];
            b.x = bbase[c4 * 16];            // xs[(c4+khalf+0)*16 + n]
            b.y = bbase[c4 * 16 + 16];       // xs[(c4+khalf+1)*16 + n]
            acc = __builtin_amdgcn_wmma_f32_16x16x4_f32(false, a, false, b,
                                                        (short)0, acc, false, false);
        }
        __syncthreads();
    }

    // ---- fused epilogue: bias, gate, relu, 1x1 conv partial ----
    const float w1p = weight1[p];
    const float* g1i = g1 + i * CC;
    const int rowAdd = 8 * (lane >> 4);
    float part = 0.f;
#pragma unroll
    for (int vv = 0; vv < 8; ++vv) {
        int cout = wav * 16 + vv + rowAdd;
        float x1 = acc[vv] + bc1[cout];
        x1 = fmaxf(x1 * w1p * g1i[cout], 0.f);
        part += x1 * W2d[cout];
    }
    red[threadIdx.x] = part;
    __syncthreads();

    if (threadIdx.x < 16) {
        const int nn = threadIdx.x;
        float tot = 0.f;
#pragma unroll
        for (int w = 0; w < 8; ++w)
            tot += red[w * 32 + nn] + red[w * 32 + nn + 16];
        const int pp  = (pt << 4) + nn;
        const int oxx = pp & 63;
        const float sig = 1.f / (1.f + expf(-(tot + b2d[0])));
        const bool msk = (v_mask[pp] > 0.f) && (oxx < valid_num[v]);
        s[((size_t)i * BB + v) * P0 + pp] = msk ? sig : -INFINITY;
    }
}

// ---------------------------------------------------------------------------
// Per-sentence reductions: scores row, argmax (first-index ties), hard
// NMS-negative, neg_loss; copies positive_map into d_out[1:].
// ---------------------------------------------------------------------------
__global__ void reduce_kernel(const float* __restrict__ s,   // [BB][BB][P0]
                              const float* __restrict__ iou, // [P0][P0]
                              const float* __restrict__ lam, // [1]
                              float* __restrict__ scores,    // [BB][BB]
                              float* __restrict__ neg_losses,// [BB]
                              float* __restrict__ pmap_out)  // [BB][P0]
{
    __shared__ float smax[256];
    __shared__ int   sidx[256];
    const int i   = blockIdx.x;
    const int tid = threadIdx.x;

    for (int v = 0; v < BB; ++v) {
        const float* sv = s + ((size_t)i * BB + v) * P0;
        float mx = -INFINITY;
        for (int p = tid; p < P0; p += 256) mx = fmaxf(mx, sv[p]);
        smax[tid] = mx;
        __syncthreads();
        for (int off = 128; off > 0; off >>= 1) {
            if (tid < off) smax[tid] = fmaxf(smax[tid], smax[tid + off]);
            __syncthreads();
        }
        if (tid == 0) scores[i * BB + v] = smax[0];
        __syncthreads();
    }

    const float* si = s + ((size_t)i * BB + i) * P0;
    float mv = -INFINITY; int mi = 0x7fffffff;
    for (int p = tid; p < P0; p += 256) {
        float vv = si[p];
        if (vv > mv || (vv == mv && p < mi)) { mv = vv; mi = p; }
    }
    smax[tid] = mv; sidx[tid] = mi;
    __syncthreads();
    for (int off = 128; off > 0; off >>= 1) {
        if (tid < off) {
            float ov = smax[tid + off]; int oi = sidx[tid + off];
            if (ov > smax[tid] || (ov == smax[tid] && oi < sidx[tid])) {
                smax[tid] = ov; sidx[tid] = oi;
            }
        }
        __syncthreads();
    }
    const int   pos_idx = sidx[0];
    const float pos_i   = smax[0];
    __syncthreads();

    for (int p = tid; p < P0; p += 256) pmap_out[(size_t)i * P0 + p] = si[p];

    const float lamv = lam[0];
    const float* iour = iou + (size_t)pos_idx * P0;
    float hm = -INFINITY;
    for (int p = tid; p < P0; p += 256) {
        float vv = si[p];
        if ((iour[p] < lamv) && (vv > -INFINITY)) hm = fmaxf(hm, vv);
    }
    smax[tid] = hm;
    __syncthreads();
    for (int off = 128; off > 0; off >>= 1) {
        if (tid < off) smax[tid] = fmaxf(smax[tid], smax[tid + off]);
        __syncthreads();
    }
    if (tid == 0)
        neg_losses[i] = fmaxf(MARGIN + smax[0] - pos_i, 0.f);
}

// ---------------------------------------------------------------------------
// Final 8x8 hinge losses + top-3 hard negatives + mean -> d_out[0]
// ---------------------------------------------------------------------------
__global__ void final_loss_kernel(const float* __restrict__ scores,
                                  const float* __restrict__ neg_losses,
                                  float* __restrict__ out)
{
    if (threadIdx.x != 0) return;
    float sc[BB][BB];
    for (int a = 0; a < BB; ++a)
        for (int b = 0; b < BB; ++b)
            sc[a][b] = scores[a * BB + b];

    float loss = 0.f;
    for (int a = 0; a < BB; ++a) {
        float row[BB];
        for (int v = 0; v < BB; ++v)
            row[v] = (v == a) ? 0.f : fmaxf(MARGIN + sc[a][v] - sc[a][a], 0.f);
        for (int t = 0; t < NEG_NUM; ++t) {
            int bi = 0; float bv = row[0];
            for (int v = 1; v < BB; ++v) if (row[v] > bv) { bv = row[v]; bi = v; }
            loss += bv / (float)BB;
            row[bi] = -1.f;
        }
    }
    for (int v = 0; v < BB; ++v) {
        float col[BB];
        for (int a = 0; a < BB; ++a)
            col[a] = (a == v) ? 0.f : fmaxf(MARGIN + sc[a][v] - sc[v][v], 0.f);
        for (int t = 0; t < NEG_NUM; ++t) {
            int bi = 0; float bv = col[0];
            for (int a = 1; a < BB; ++a) if (col[a] > bv) { bv = col[a]; bi = a; }
            loss += bv / (float)BB;
            col[bi] = -1.f;
        }
    }
    float nl = 0.f;
    for (int a = 0; a < BB; ++a) nl += neg_losses[a];
    out[0] = loss + nl / (float)BB;
}

// ---------------------------------------------------------------------------
extern "C" void kernel_launch(void* const* d_in, const int* in_sizes, int n_in,
                              void* d_out, int out_size, void* d_ws, size_t ws_size,
                              hipStream_t stream)
{
    const float* v_map     = (const float*)d_in[0];
    const float* words     = (const float*)d_in[1];
    const float* w_masks   = (const float*)d_in[2];
    const float* v_mask    = (const float*)d_in[3];
    const int*   valid_num = (const int*)  d_in[4];
    const float* iou       = (const float*)d_in[5];
    const float* lam       = (const float*)d_in[6];
    const float* Wc0       = (const float*)d_in[7];
    const float* bc0       = (const float*)d_in[8];
    const float* Wc1       = (const float*)d_in[9];
    const float* bc1       = (const float*)d_in[10];
    const float* Wp0       = (const float*)d_in[11];
    const float* bp0       = (const float*)d_in[12];
    const float* Wp1       = (const float*)d_in[13];
    const float* bp1       = (const float*)d_in[14];
    const float* W2d       = (const float*)d_in[15];
    const float* b2d       = (const float*)d_in[16];

    float* ws = (float*)d_ws;
    size_t off = 0;
    float* conv0   = ws + off; off += (size_t)BB * CC * P1;  // 17.8 MB
    float* Wt0     = ws + off; off += (size_t)CC * KDIM;     // 0.6 MB
    float* Wt1     = ws + off; off += (size_t)CC * KDIM;     // 0.6 MB
    float* weight0 = ws + off; off += P1;
    float* weight1 = ws + off; off += P0;
    float* g0      = ws + off; off += BB * CC;
    float* g1      = ws + off; off += BB * CC;
    float* sbuf    = ws + off; off += (size_t)BB * BB * P0;  // 1 MB
    float* scores  = ws + off; off += BB * BB;
    float* neg_l   = ws + off; off += BB;

    float* out = (float*)d_out;           // [0] = loss, [1..] = positive_map

    const int nTW = (CC * KDIM + 255) / 256;
    transpose_w_kernel<<<nTW, 256, 0, stream>>>(Wc0, Wt0);
    transpose_w_kernel<<<nTW, 256, 0, stream>>>(Wc1, Wt1);

    prep_kernel<<<1, 256, 0, stream>>>(v_mask, words, w_masks, Wp0, bp0, Wp1, bp1,
                                       weight0, weight1, g0, g1);

    conv0_kernel<<<BB * ((P1 + 15) / 16), 256, 0, stream>>>(v_map, Wt0, bc0, conv0);

    conv1_score_kernel<<<BB * BB * (P0 / 16), 256, 0, stream>>>(
        conv0, Wt1, bc1, weight0, weight1, g0, g1, W2d, b2d,
        v_mask, valid_num, sbuf);

    reduce_kernel<<<BB, 256, 0, stream>>>(sbuf, iou, lam, scores, neg_l, out + 1);

    final_loss_kernel<<<1, 32, 0, stream>>>(scores, neg_l, out);
}